// point_backbone_4587025072298
// MI455X (gfx1250) — compile-verified
//
#include <hip/hip_runtime.h>
#include <hip/hip_bf16.h>

typedef __attribute__((ext_vector_type(2))) float v2f;
typedef __attribute__((ext_vector_type(8))) float v8f;

#define BB 4
#define NN 2048
#define KNBR 20
#define SLOPE 0.2f
#define EPS 1e-5f
#define ROWTILES ((BB * NN) / 16)   /* 512 */
#define ROWG (ROWTILES / 8)         /* 64 row-tile groups of 8 waves */

// ---------------------------------------------------------------------------
// transpose x (B,3,N) -> x_t (B*N,3)
__global__ void transpose_x_kernel(const float* __restrict__ x, float* __restrict__ xt) {
    int t = blockIdx.x * blockDim.x + threadIdx.x;
    if (t >= BB * NN) return;
    int b = t / NN, n = t % NN;
    #pragma unroll
    for (int c = 0; c < 3; ++c)
        xt[(size_t)t * 3 + c] = x[((size_t)b * 3 + c) * NN + n];
}

// squared norms per row
__global__ void sqnorm_kernel(const float* __restrict__ feat, float* __restrict__ xx, int C) {
    int t = blockIdx.x * blockDim.x + threadIdx.x;
    if (t >= BB * NN) return;
    const float* f = feat + (size_t)t * C;
    float s = 0.f;
    for (int c = 0; c < C; ++c) s += f[c] * f[c];
    xx[t] = s;
}

// ---------------------------------------------------------------------------
// KNN: one workgroup (256 thr) per (b,i). pd = 2*dot - xx_i - xx_j, top-20,
// tie-break lowest index (matches jax.lax.top_k).
__global__ void knn_kernel(const float* __restrict__ feat, const float* __restrict__ xx,
                           int* __restrict__ idx, int C) {
    __shared__ float dist[NN];
    __shared__ float fi[128];
    __shared__ float rv[256];
    __shared__ int   ri[256];
    int bi = blockIdx.x;           // b*N + i
    int b = bi / NN;
    int t = threadIdx.x;
    if (t < C) fi[t] = feat[(size_t)bi * C + t];
    __syncthreads();
    float xi = xx[bi];
    for (int j = t; j < NN; j += 256) {
        const float* fj = feat + ((size_t)(b * NN + j)) * C;
        float d = 0.f;
        for (int c = 0; c < C; ++c) d += fi[c] * fj[c];
        dist[j] = 2.0f * d - xi - xx[b * NN + j];
    }
    __syncthreads();
    for (int kk = 0; kk < KNBR; ++kk) {
        float bv = -__builtin_inff(); int bj = NN;
        for (int j = t; j < NN; j += 256) {
            float v = dist[j];
            if (v > bv) { bv = v; bj = j; }
        }
        rv[t] = bv; ri[t] = bj;
        __syncthreads();
        for (int st = 128; st > 0; st >>= 1) {
            if (t < st) {
                float ov = rv[t + st]; int oj = ri[t + st];
                if (ov > rv[t] || (ov == rv[t] && oj < ri[t])) { rv[t] = ov; ri[t] = oj; }
            }
            __syncthreads();
        }
        if (t == 0) { idx[(size_t)bi * KNBR + kk] = ri[0]; dist[ri[0]] = -__builtin_inff(); }
        __syncthreads();
    }
}

// ---------------------------------------------------------------------------
// Generic branchless edge-feature element (used for layer 1, C=3 odd + pad):
__device__ __forceinline__ float loadE(const float* fc, const float* fn, int C, int C2, int c) {
    int cm = (c < C) ? c : (c - C);
    cm = (cm < C) ? cm : 0;                 // clamp for layer-1 K padding
    float ctr = fc[cm];
    float nb  = fn[cm];
    float v = (c < C) ? (nb - ctr) : ctr;
    return (c < C2) ? v : 0.f;
}

// A-fragment pair: EVEN path does two contiguous v2f loads + one select
// (valid when C is even: a (cA,cA+1) pair never straddles the nbr/ctr split).
template<bool EVEN>
__device__ __forceinline__ v2f loadA(const float* fc, const float* fn, int C, int C2, int cA) {
    v2f a;
    if (EVEN) {
        bool low = cA < C;
        int cm = low ? cA : cA - C;
        v2f ctr = *(const v2f*)(fc + cm);
        v2f nb  = *(const v2f*)(fn + cm);
        a.x = low ? (nb.x - ctr.x) : ctr.x;
        a.y = low ? (nb.y - ctr.y) : ctr.y;
    } else {
        a.x = loadE(fc, fn, C, C2, cA);
        a.y = loadE(fc, fn, C, C2, cA + 1);
    }
    return a;
}

// ---------------------------------------------------------------------------
// EdgeConv pass 1: per-channel sum / sumsq of h = e @ W^T over all (b,n,k).
// Block = 8 waves sharing a 32-col W tile in LDS; each wave computes a 16x32
// tile (2 WMMA accumulators share each A-fragment).
template<bool EVEN>
__global__ __launch_bounds__(256)
void edgeconv_stats_kernel(const float* __restrict__ feat, const int* __restrict__ idx,
                           const float* __restrict__ W,
                           float* __restrict__ sums, float* __restrict__ sumsq,
                           int C, int O) {
    __shared__ float Wt[256 * 32];              // K2 <= 256, 32 cols (32 KB)
    const int C2 = 2 * C;
    const int K2 = (C2 + 3) & ~3;
    int ct = blockIdx.x / ROWG;                 // 32-col tile (shared by block)
    int rg = blockIdx.x % ROWG;                 // row-tile group
    for (int i = threadIdx.x; i < K2 * 32; i += 256) {
        int c = i >> 5, l = i & 31;
        Wt[i] = (c < C2) ? W[(size_t)(ct * 32 + l) * C2 + c] : 0.f;
    }
    __syncthreads();

    int w = threadIdx.x >> 5;
    int rt = rg * 8 + w;
    int lane = threadIdx.x & 31;
    int l16 = lane & 15, hi = lane >> 4;
    int rowA = rt * 16 + l16;
    int bA = rowA / NN;
    int col0 = ct * 32 + l16;
    int col1 = col0 + 16;
    const float* fc = feat + (size_t)rowA * C;
    float s0 = 0.f, q0 = 0.f, s1 = 0.f, q1 = 0.f;
    for (int k = 0; k < KNBR; ++k) {
        int nbr = idx[(size_t)rowA * KNBR + k];
        const float* fn = feat + ((size_t)(bA * NN + nbr)) * C;
        v8f acc0 = {0.f,0.f,0.f,0.f,0.f,0.f,0.f,0.f};
        v8f acc1 = {0.f,0.f,0.f,0.f,0.f,0.f,0.f,0.f};
        for (int c0 = 0; c0 < K2; c0 += 4) {
            int cA = c0 + hi * 2;
            v2f a = loadA<EVEN>(fc, fn, C, C2, cA);
            v2f b0, b1;
            b0.x = Wt[cA * 32 + l16];        b0.y = Wt[(cA + 1) * 32 + l16];
            b1.x = Wt[cA * 32 + 16 + l16];   b1.y = Wt[(cA + 1) * 32 + 16 + l16];
            acc0 = __builtin_amdgcn_wmma_f32_16x16x4_f32(false, a, false, b0,
                                                         (short)0, acc0, false, false);
            acc1 = __builtin_amdgcn_wmma_f32_16x16x4_f32(false, a, false, b1,
                                                         (short)0, acc1, false, false);
        }
        #pragma unroll
        for (int j = 0; j < 8; ++j) {
            float v0 = acc0[j]; s0 += v0; q0 += v0 * v0;
            float v1 = acc1[j]; s1 += v1; q1 += v1 * v1;
        }
    }
    s0 += __shfl_down(s0, 16, 32);
    q0 += __shfl_down(q0, 16, 32);
    s1 += __shfl_down(s1, 16, 32);
    q1 += __shfl_down(q1, 16, 32);
    if (hi == 0) {
        atomicAdd(&sums[col0], s0);
        atomicAdd(&sumsq[col0], q0);
        atomicAdd(&sums[col1], s1);
        atomicAdd(&sumsq[col1], q1);
    }
}

// per-channel BN affine from accumulated sums
__global__ void bn_finalize_kernel(const float* __restrict__ sums, const float* __restrict__ sumsq,
                                   const float* __restrict__ g, const float* __restrict__ bta,
                                   float* __restrict__ scale, float* __restrict__ shift,
                                   int O, float invcnt) {
    int o = blockIdx.x * blockDim.x + threadIdx.x;
    if (o >= O) return;
    float m = sums[o] * invcnt;
    float v = sumsq[o] * invcnt - m * m;
    float sc = g[o] * rsqrtf(v + EPS);
    scale[o] = sc;
    shift[o] = bta[o] - m * sc;
}

// EdgeConv pass 2: recompute GEMM, apply BN + leaky-relu, max over k, store.
template<bool EVEN>
__global__ __launch_bounds__(256)
void edgeconv_apply_kernel(const float* __restrict__ feat, const int* __restrict__ idx,
                           const float* __restrict__ W,
                           const float* __restrict__ scale, const float* __restrict__ shift,
                           float* __restrict__ outp, int C, int O) {
    __shared__ float Wt[256 * 32];
    const int C2 = 2 * C;
    const int K2 = (C2 + 3) & ~3;
    int ct = blockIdx.x / ROWG;
    int rg = blockIdx.x % ROWG;
    for (int i = threadIdx.x; i < K2 * 32; i += 256) {
        int c = i >> 5, l = i & 31;
        Wt[i] = (c < C2) ? W[(size_t)(ct * 32 + l) * C2 + c] : 0.f;
    }
    __syncthreads();

    int w = threadIdx.x >> 5;
    int rt = rg * 8 + w;
    int lane = threadIdx.x & 31;
    int l16 = lane & 15, hi = lane >> 4;
    int rowA = rt * 16 + l16;
    int bA = rowA / NN;
    int col0 = ct * 32 + l16;
    int col1 = col0 + 16;
    const float* fc = feat + (size_t)rowA * C;
    float sc0 = scale[col0], sh0 = shift[col0];
    float sc1 = scale[col1], sh1 = shift[col1];
    float best0[8], best1[8];
    #pragma unroll
    for (int j = 0; j < 8; ++j) { best0[j] = -__builtin_inff(); best1[j] = -__builtin_inff(); }
    for (int k = 0; k < KNBR; ++k) {
        int nbr = idx[(size_t)rowA * KNBR + k];
        const float* fn = feat + ((size_t)(bA * NN + nbr)) * C;
        v8f acc0 = {0.f,0.f,0.f,0.f,0.f,0.f,0.f,0.f};
        v8f acc1 = {0.f,0.f,0.f,0.f,0.f,0.f,0.f,0.f};
        for (int c0 = 0; c0 < K2; c0 += 4) {
            int cA = c0 + hi * 2;
            v2f a = loadA<EVEN>(fc, fn, C, C2, cA);
            v2f b0, b1;
            b0.x = Wt[cA * 32 + l16];        b0.y = Wt[(cA + 1) * 32 + l16];
            b1.x = Wt[cA * 32 + 16 + l16];   b1.y = Wt[(cA + 1) * 32 + 16 + l16];
            acc0 = __builtin_amdgcn_wmma_f32_16x16x4_f32(false, a, false, b0,
                                                         (short)0, acc0, false, false);
            acc1 = __builtin_amdgcn_wmma_f32_16x16x4_f32(false, a, false, b1,
                                                         (short)0, acc1, false, false);
        }
        #pragma unroll
        for (int j = 0; j < 8; ++j) {
            float v0 = acc0[j] * sc0 + sh0; v0 = (v0 >= 0.f) ? v0 : SLOPE * v0;
            best0[j] = fmaxf(best0[j], v0);
            float v1 = acc1[j] * sc1 + sh1; v1 = (v1 >= 0.f) ? v1 : SLOPE * v1;
            best1[j] = fmaxf(best1[j], v1);
        }
    }
    // C/D layout: lanes 0-15 rows M=0..7, lanes 16-31 rows M=8..15; N = lane%16
    #pragma unroll
    for (int j = 0; j < 8; ++j) {
        int r = rt * 16 + hi * 8 + j;
        outp[(size_t)r * O + col0] = best0[j];
        outp[(size_t)r * O + col1] = best1[j];
    }
}

// ---------------------------------------------------------------------------
// materialize xc = concat(x1,x2,x3,x4)  (B*N, 512)
__global__ void concat_kernel(const float* __restrict__ x1, const float* __restrict__ x2,
                              const float* __restrict__ x3, const float* __restrict__ x4,
                              float* __restrict__ xc) {
    size_t i = (size_t)blockIdx.x * blockDim.x + threadIdx.x;
    if (i >= (size_t)BB * NN * 512) return;
    int bn = (int)(i >> 9);
    int c = (int)(i & 511);
    float v;
    if (c < 64)       v = x1[(size_t)bn * 64  + c];
    else if (c < 128) v = x2[(size_t)bn * 64  + (c - 64)];
    else if (c < 256) v = x3[(size_t)bn * 128 + (c - 128)];
    else              v = x4[(size_t)bn * 256 + (c - 256)];
    xc[i] = v;
}

// h5 = xc @ W5^T  (8192 x 1024, K=512); 32-col W5 tile in LDS, 2 accumulators
__global__ __launch_bounds__(256)
void gemm_h5_kernel(const float* __restrict__ xc, const float* __restrict__ W5,
                    float* __restrict__ h5) {
    __shared__ float Wt[512 * 32];              // 64 KB
    int ct = blockIdx.x / ROWG;
    int rg = blockIdx.x % ROWG;
    for (int i = threadIdx.x; i < 512 * 32; i += 256) {
        int c = i >> 5, l = i & 31;
        Wt[i] = W5[(size_t)(ct * 32 + l) * 512 + c];
    }
    __syncthreads();

    int w = threadIdx.x >> 5;
    int rt = rg * 8 + w;
    int lane = threadIdx.x & 31;
    int l16 = lane & 15, hi = lane >> 4;
    int rowA = rt * 16 + l16;
    int col0 = ct * 32 + l16;
    int col1 = col0 + 16;
    const float* arow = xc + (size_t)rowA * 512;
    v8f acc0 = {0.f,0.f,0.f,0.f,0.f,0.f,0.f,0.f};
    v8f acc1 = {0.f,0.f,0.f,0.f,0.f,0.f,0.f,0.f};
    for (int c0 = 0; c0 < 512; c0 += 4) {
        int cA = c0 + hi * 2;
        v2f a = *(const v2f*)(arow + cA);        // contiguous pair -> b64 load
        v2f b0, b1;
        b0.x = Wt[cA * 32 + l16];        b0.y = Wt[(cA + 1) * 32 + l16];
        b1.x = Wt[cA * 32 + 16 + l16];   b1.y = Wt[(cA + 1) * 32 + 16 + l16];
        acc0 = __builtin_amdgcn_wmma_f32_16x16x4_f32(false, a, false, b0,
                                                     (short)0, acc0, false, false);
        acc1 = __builtin_amdgcn_wmma_f32_16x16x4_f32(false, a, false, b1,
                                                     (short)0, acc1, false, false);
    }
    #pragma unroll
    for (int j = 0; j < 8; ++j) {
        int r = rt * 16 + hi * 8 + j;
        h5[(size_t)r * 1024 + col0] = acc0[j];
        h5[(size_t)r * 1024 + col1] = acc1[j];
    }
}

// deterministic per-channel stats over rows of h5 (BN over axes (0,1))
__global__ void col_stats_kernel(const float* __restrict__ h,
                                 const float* __restrict__ g, const float* __restrict__ bta,
                                 float* __restrict__ scale, float* __restrict__ shift) {
    const int rows = BB * NN, F = 1024;
    int f = blockIdx.x;
    float s = 0.f, s2 = 0.f;
    for (int r = threadIdx.x; r < rows; r += blockDim.x) {
        float v = h[(size_t)r * F + f];
        s += v; s2 += v * v;
    }
    __shared__ float rs[256], rs2[256];
    rs[threadIdx.x] = s; rs2[threadIdx.x] = s2;
    __syncthreads();
    for (int st = 128; st > 0; st >>= 1) {
        if (threadIdx.x < st) {
            rs[threadIdx.x]  += rs[threadIdx.x + st];
            rs2[threadIdx.x] += rs2[threadIdx.x + st];
        }
        __syncthreads();
    }
    if (threadIdx.x == 0) {
        float m = rs[0] / rows;
        float v = rs2[0] / rows - m * m;
        float sc = g[f] * rsqrtf(v + EPS);
        scale[f] = sc;
        shift[f] = bta[f] - m * sc;
    }
}

__global__ void bn_lrelu_kernel(float* __restrict__ h, const float* __restrict__ scale,
                                const float* __restrict__ shift) {
    size_t i = (size_t)blockIdx.x * blockDim.x + threadIdx.x;
    if (i >= (size_t)BB * NN * 1024) return;
    int f = (int)(i & 1023);
    float v = h[i] * scale[f] + shift[f];
    h[i] = (v >= 0.f) ? v : SLOPE * v;
}

// ---------------------------------------------------------------------------
// pooling bins: for nb in {1,2,4,8,16,32}: max + mean over segments -> (B,63,1024)
__global__ void pool_kernel(const float* __restrict__ h, float* __restrict__ bins) {
    const int F = 1024;
    int b = blockIdx.x / 63, s = blockIdx.x % 63;
    int L = 31 - __clz(s + 1);        // level (nb = 1<<L)
    int j = s - ((1 << L) - 1);
    int len = NN >> L;
    int start = j * len;
    for (int f = threadIdx.x; f < F; f += blockDim.x) {
        const float* p = h + ((size_t)(b * NN + start)) * F + f;
        float mx = -__builtin_inff(), sm = 0.f;
        for (int r = 0; r < len; ++r) {
            float v = p[(size_t)r * F];
            mx = fmaxf(mx, v); sm += v;
        }
        bins[((size_t)(b * 63 + s)) * F + f] = mx + sm / (float)len;
    }
}

// out[s,b,o] = bins[b,s,:] . Wf[o,:] + bfeat[o]   (63,4,256)
__global__ void final_proj_kernel(const float* __restrict__ bins, const float* __restrict__ Wf,
                                  const float* __restrict__ bfeat, float* __restrict__ out) {
    int t = blockIdx.x * blockDim.x + threadIdx.x;
    if (t >= 63 * 4 * 256) return;
    int o = t & 255;
    int b = (t >> 8) & 3;
    int s = t >> 10;
    const float* br = bins + ((size_t)(b * 63 + s)) * 1024;
    const float* wr = Wf + (size_t)o * 1024;
    float acc = bfeat[o];
    for (int f = 0; f < 1024; ++f) acc += br[f] * wr[f];
    out[t] = acc;   // t == (s*4+b)*256 + o
}

// ---------------------------------------------------------------------------
extern "C" void kernel_launch(void* const* d_in, const int* in_sizes, int n_in,
                              void* d_out, int out_size, void* d_ws, size_t ws_size,
                              hipStream_t stream) {
    const float* x  = (const float*)d_in[0];
    const float* W1 = (const float*)d_in[1];
    const float* g1 = (const float*)d_in[2];
    const float* b1 = (const float*)d_in[3];
    const float* W2 = (const float*)d_in[4];
    const float* g2 = (const float*)d_in[5];
    const float* b2 = (const float*)d_in[6];
    const float* W3 = (const float*)d_in[7];
    const float* g3 = (const float*)d_in[8];
    const float* b3 = (const float*)d_in[9];
    const float* W4 = (const float*)d_in[10];
    const float* g4 = (const float*)d_in[11];
    const float* b4 = (const float*)d_in[12];
    const float* W5 = (const float*)d_in[13];
    const float* g5 = (const float*)d_in[14];
    const float* b5 = (const float*)d_in[15];
    const float* Wf = (const float*)d_in[16];
    const float* bf = (const float*)d_in[17];
    float* out = (float*)d_out;

    const int BN = BB * NN;   // 8192
    float* ws = (float*)d_ws;
    size_t off = 0;
    float* x_t   = ws + off; off += (size_t)BN * 3;
    float* xx    = ws + off; off += BN;
    int*   idx   = (int*)(ws + off); off += (size_t)BN * KNBR;
    float* x1    = ws + off; off += (size_t)BN * 64;
    float* x2    = ws + off; off += (size_t)BN * 64;
    float* x3    = ws + off; off += (size_t)BN * 128;
    float* x4    = ws + off; off += (size_t)BN * 256;
    float* xc    = ws + off; off += (size_t)BN * 512;
    float* h5    = ws + off; off += (size_t)BN * 1024;
    float* sums  = ws + off; off += 1024;
    float* sumsq = ws + off; off += 1024;
    float* scale = ws + off; off += 1024;
    float* shift = ws + off; off += 1024;
    float* bins  = ws + off; off += (size_t)BB * 63 * 1024;

    transpose_x_kernel<<<(BN + 255) / 256, 256, 0, stream>>>(x, x_t);

    auto run_edgeconv = [&](const float* feat, int C, int O, const float* W,
                            const float* g, const float* bta, float* outb) {
        sqnorm_kernel<<<(BN + 255) / 256, 256, 0, stream>>>(feat, xx, C);
        knn_kernel<<<BN, 256, 0, stream>>>(feat, xx, idx, C);
        hipMemsetAsync(sums, 0, 2 * 1024 * sizeof(float), stream);  // sums + sumsq
        int blocks = (O / 32) * ROWG;           // one block per (32-col tile, 8-row group)
        if ((C & 1) == 0) {
            edgeconv_stats_kernel<true><<<blocks, 256, 0, stream>>>(feat, idx, W, sums, sumsq, C, O);
        } else {
            edgeconv_stats_kernel<false><<<blocks, 256, 0, stream>>>(feat, idx, W, sums, sumsq, C, O);
        }
        bn_finalize_kernel<<<(O + 255) / 256, 256, 0, stream>>>(
            sums, sumsq, g, bta, scale, shift, O, 1.0f / ((float)BN * KNBR));
        if ((C & 1) == 0) {
            edgeconv_apply_kernel<true><<<blocks, 256, 0, stream>>>(feat, idx, W, scale, shift, outb, C, O);
        } else {
            edgeconv_apply_kernel<false><<<blocks, 256, 0, stream>>>(feat, idx, W, scale, shift, outb, C, O);
        }
    };

    run_edgeconv(x_t, 3,   64,  W1, g1, b1, x1);
    run_edgeconv(x1,  64,  64,  W2, g2, b2, x2);
    run_edgeconv(x2,  64,  128, W3, g3, b3, x3);
    run_edgeconv(x3,  128, 256, W4, g4, b4, x4);

    {
        size_t total = (size_t)BN * 512;
        concat_kernel<<<(unsigned)((total + 255) / 256), 256, 0, stream>>>(x1, x2, x3, x4, xc);
        int blocks = (1024 / 32) * ROWG;
        gemm_h5_kernel<<<blocks, 256, 0, stream>>>(xc, W5, h5);
    }
    col_stats_kernel<<<1024, 256, 0, stream>>>(h5, g5, b5, scale, shift);
    {
        size_t total = (size_t)BN * 1024;
        bn_lrelu_kernel<<<(unsigned)((total + 255) / 256), 256, 0, stream>>>(h5, scale, shift);
    }
    pool_kernel<<<BB * 63, 256, 0, stream>>>(h5, bins);
    final_proj_kernel<<<(63 * 4 * 256 + 255) / 256, 256, 0, stream>>>(bins, Wf, bf, out);
}